// PatchROIPooler_60739427500698
// MI455X (gfx1250) — compile-verified
//
#include <hip/hip_runtime.h>
#include <hip/hip_bf16.h>

#define N_SIDE   14
#define N_PATCH  196
#define D_BB     384
#define D_MODEL  512
#define BT_TOTAL 2048
#define LN_EPS   1e-5f

#define A_STRIDE 388   // 384 + 4 dword pad: (m*388 + k) % 64 = (4m + k) % 64 -> conflict-free
#define O_STRIDE 516   // 512 + 4 dword pad: breaks half-wave (row+8) bank aliasing

typedef __attribute__((ext_vector_type(2))) float v2f;
typedef __attribute__((ext_vector_type(8))) float v8f;

// ---------------------------------------------------------------------------
// Kernel 1: ROI mean pool.  One block per (b,t) row; 128 threads x 3 channels.
// Reads only bbox-covered patches (~1/9 of the 616 MB input on average).
// ---------------------------------------------------------------------------
__global__ __launch_bounds__(128) void roi_pool_kernel(
    const float* __restrict__ pt,      // [BT, 196, 384]
    const float* __restrict__ bboxes,  // [BT, 4]
    float* __restrict__ pooled)        // [BT, 384]
{
    const int bt  = blockIdx.x;
    const int tid = threadIdx.x;

    const float bx1 = bboxes[bt * 4 + 0];
    const float by1 = bboxes[bt * 4 + 1];
    const float bx2 = bboxes[bt * 4 + 2];
    const float by2 = bboxes[bt * 4 + 3];

    // Mirror the JAX semantics:
    // x1 = clip(b*14, 0, 13) truncated; x2 = clip(ceil(clip(b*14,0,14)), 1, 14)
    int x1 = (int)fminf(fmaxf(bx1 * 14.0f, 0.0f), 13.0f);
    int y1 = (int)fminf(fmaxf(by1 * 14.0f, 0.0f), 13.0f);
    int x2 = (int)fminf(fmaxf(ceilf(fminf(fmaxf(bx2 * 14.0f, 0.0f), 14.0f)), 1.0f), 14.0f);
    int y2 = (int)fminf(fmaxf(ceilf(fminf(fmaxf(by2 * 14.0f, 0.0f), 14.0f)), 1.0f), 14.0f);
    x2 = max(x2, x1 + 1);
    y2 = max(y2, y1 + 1);

    const float inv_cnt = 1.0f / (float)((y2 - y1) * (x2 - x1));

    float a0 = 0.0f, a1 = 0.0f, a2 = 0.0f;
    for (int y = y1; y < y2; ++y) {
        for (int x = x1; x < x2; ++x) {
            const float* p = pt + ((size_t)bt * N_PATCH + y * N_SIDE + x) * D_BB;
            a0 += p[tid];
            a1 += p[tid + 128];
            a2 += p[tid + 256];
        }
    }
    float* o = pooled + (size_t)bt * D_BB;
    o[tid]       = a0 * inv_cnt;
    o[tid + 128] = a1 * inv_cnt;
    o[tid + 256] = a2 * inv_cnt;
}

// ---------------------------------------------------------------------------
// Kernel 2: (pooled @ W + b) -> LayerNorm -> visibility blend, WMMA f32.
// 128 blocks; each block owns 16 rows x 512 cols (full LN rows stay local).
// 8 wave32s; wave w owns cols [w*64, w*64+64) = 4 accumulator tiles.
// 96 K-steps of v_wmma_f32_16x16x4_f32 per tile.
// ---------------------------------------------------------------------------
__global__ __launch_bounds__(256) void gemm_ln_kernel(
    const float* __restrict__ pooled,      // [BT, 384]
    const float* __restrict__ Wmat,        // [384, 512]
    const float* __restrict__ bias,        // [512]
    const float* __restrict__ gamma,       // [512]
    const float* __restrict__ beta,        // [512]
    const float* __restrict__ mask_token,  // [512]
    const float* __restrict__ obj_visible, // [BT]
    float* __restrict__ out)               // [BT, 512]
{
    __shared__ float Alds[16 * A_STRIDE];
    __shared__ float Olds[16 * O_STRIDE];

    const int tid  = threadIdx.x;
    const int row0 = blockIdx.x * 16;

    // ---- Stage A tile (16 x 384 f32) into padded LDS via float4 ----
    for (int q = tid; q < 16 * (D_BB / 4); q += 256) {
        const int m  = q / (D_BB / 4);
        const int kq = q % (D_BB / 4);
        float4 v = ((const float4*)(pooled + (size_t)(row0 + m) * D_BB))[kq];
        *(float4*)&Alds[m * A_STRIDE + kq * 4] = v;
    }
    __syncthreads();

    const int wave    = tid >> 5;
    const int lane    = tid & 31;
    const int half    = lane >> 4;   // 0: lanes 0-15, 1: lanes 16-31
    const int l15     = lane & 15;
    const int colbase = wave * 64;

    // ISA 7.12.2 fragment layouts for 16x16x4 f32:
    //  A: lane (M=l15), VGPR0/1 hold K = half*2, half*2+1
    //  B: lane (N=l15), VGPR0/1 hold K = half*2, half*2+1
    const int kb = half * 2;

    v8f c0 = {}, c1 = {}, c2 = {}, c3 = {};

    for (int k0 = 0; k0 < D_BB; k0 += 4) {
        v2f a = *(const v2f*)&Alds[l15 * A_STRIDE + k0 + kb];

        const float* wrow = Wmat + (size_t)(k0 + kb) * D_MODEL + colbase + l15;
        v2f b0, b1, b2, b3;
        b0.x = wrow[0];   b0.y = wrow[D_MODEL];
        b1.x = wrow[16];  b1.y = wrow[16 + D_MODEL];
        b2.x = wrow[32];  b2.y = wrow[32 + D_MODEL];
        b3.x = wrow[48];  b3.y = wrow[48 + D_MODEL];

        c0 = __builtin_amdgcn_wmma_f32_16x16x4_f32(false, a, false, b0, (short)0, c0, false, false);
        c1 = __builtin_amdgcn_wmma_f32_16x16x4_f32(false, a, false, b1, (short)0, c1, false, false);
        c2 = __builtin_amdgcn_wmma_f32_16x16x4_f32(false, a, false, b2, (short)0, c2, false, false);
        c3 = __builtin_amdgcn_wmma_f32_16x16x4_f32(false, a, false, b3, (short)0, c3, false, false);
    }

    // ---- Spill C tiles (+bias) to padded LDS: VGPR r -> row (r + 8*half) ----
    {
        v8f acc[4] = {c0, c1, c2, c3};
#pragma unroll
        for (int t = 0; t < 4; ++t) {
            const int col = colbase + t * 16 + l15;
            const float bv = bias[col];
#pragma unroll
            for (int r = 0; r < 8; ++r) {
                const int m = r + half * 8;
                Olds[m * O_STRIDE + col] = acc[t][r] + bv;
            }
        }
    }
    __syncthreads();

    // ---- LayerNorm + mask-token blend: wave w handles rows 2w, 2w+1 ----
#pragma unroll
    for (int rr = 0; rr < 2; ++rr) {
        const int row = wave * 2 + rr;
        const int rg  = row0 + row;
        const float* rptr = &Olds[row * O_STRIDE];

        float s = 0.0f;
        for (int j = lane; j < D_MODEL; j += 32) s += rptr[j];
#pragma unroll
        for (int off = 16; off > 0; off >>= 1) s += __shfl_xor(s, off, 32);
        const float mu = s * (1.0f / (float)D_MODEL);

        float vs = 0.0f;
        for (int j = lane; j < D_MODEL; j += 32) {
            const float d = rptr[j] - mu;
            vs += d * d;
        }
#pragma unroll
        for (int off = 16; off > 0; off >>= 1) vs += __shfl_xor(vs, off, 32);
        const float rstd = rsqrtf(vs * (1.0f / (float)D_MODEL) + LN_EPS);

        const float vis = obj_visible[rg];
        float* orow = out + (size_t)rg * D_MODEL;
        for (int j = lane; j < D_MODEL; j += 32) {
            const float val = (rptr[j] - mu) * rstd * gamma[j] + beta[j];
            orow[j] = vis * val + (1.0f - vis) * mask_token[j];
        }
    }
}

// ---------------------------------------------------------------------------
extern "C" void kernel_launch(void* const* d_in, const int* in_sizes, int n_in,
                              void* d_out, int out_size, void* d_ws, size_t ws_size,
                              hipStream_t stream) {
    const float* patch_tokens = (const float*)d_in[0]; // [2048,196,384]
    const float* bboxes       = (const float*)d_in[1]; // [2048,4]
    const float* obj_visible  = (const float*)d_in[2]; // [2048]
    // d_in[3] = B (int scalar), d_in[4] = T (int scalar) -- unused (output is flat)
    const float* Wmat         = (const float*)d_in[5]; // [384,512]
    const float* bias         = (const float*)d_in[6]; // [512]
    const float* gamma        = (const float*)d_in[7]; // [512]
    const float* beta         = (const float*)d_in[8]; // [512]
    const float* mask_token   = (const float*)d_in[9]; // [1,512]
    float* out    = (float*)d_out;                     // [2048,512]
    float* pooled = (float*)d_ws;                      // 2048*384 f32 = 3 MB scratch

    roi_pool_kernel<<<BT_TOTAL, 128, 0, stream>>>(patch_tokens, bboxes, pooled);
    gemm_ln_kernel<<<BT_TOTAL / 16, 256, 0, stream>>>(pooled, Wmat, bias, gamma, beta,
                                                      mask_token, obj_visible, out);
}